// GCN_10943576670613
// MI455X (gfx1250) — compile-verified
//
#include <hip/hip_runtime.h>
#include <hip/hip_bf16.h>

typedef __attribute__((ext_vector_type(2))) float v2f;
typedef __attribute__((ext_vector_type(8))) float v8f;

#define CH 128          // hidden channels (H)
#define BN_EPS 1e-5f

// Hardware fp32 atomic add (global_atomic_add_f32, non-returning -> STOREcnt).
__device__ __forceinline__ void atomicAddF(float* p, float v) {
    (void)__hip_atomic_fetch_add(p, v, __ATOMIC_RELAXED, __HIP_MEMORY_SCOPE_AGENT);
}

// ---------------- degree / normalization ----------------
__global__ void gcn_deg_init(float* deg, int n) {
    int i = blockIdx.x * blockDim.x + threadIdx.x;
    if (i < n) deg[i] = 1.0f;                 // self loop
}
__global__ void gcn_deg_count(const int* __restrict__ dst, float* deg, int e) {
    int i = blockIdx.x * blockDim.x + threadIdx.x;
    if (i < e) atomicAddF(&deg[dst[i]], 1.0f);
}
__global__ void gcn_deg_fin(float* deg, int n) {
    int i = blockIdx.x * blockDim.x + threadIdx.x;
    if (i < n) deg[i] = rsqrtf(deg[i]);       // deg >= 1 always
}
__global__ void gcn_zero(float* p, int n) {
    int i = blockIdx.x * blockDim.x + threadIdx.x;
    if (i < n) p[i] = 0.0f;
}

// ---------------- fp32 WMMA GEMM:  C[M x 128] = A[M x K] @ W[K x 128] ----------------
// One wave computes a 16-row strip across all 8 column tiles (A fragment reused 8x).
__global__ void gcn_gemm_wmma(const float* __restrict__ A, const float* __restrict__ W,
                              float* __restrict__ C, int M, int K) {
    int wave = blockIdx.x * (blockDim.x >> 5) + (threadIdx.x >> 5);
    int lane = threadIdx.x & 31;
    if (wave * 16 >= M) return;

    int mrow  = wave * 16 + (lane & 15);
    int khalf = (lane >> 4) << 1;             // 0 for lanes 0-15, 2 for lanes 16-31
    int ncol  = lane & 15;

    v8f acc[8];
#pragma unroll
    for (int t = 0; t < 8; ++t) acc[t] = (v8f){0,0,0,0,0,0,0,0};

    for (int k0 = 0; k0 < K; k0 += 4) {
        v2f a;
        const float* ar = A + (long)mrow * K + k0 + khalf;
        a.x = ar[0];
        a.y = ar[1];
        const float* w0 = W + (long)(k0 + khalf) * CH;
        const float* w1 = w0 + CH;
#pragma unroll
        for (int t = 0; t < 8; ++t) {
            v2f b;
            b.x = w0[t * 16 + ncol];
            b.y = w1[t * 16 + ncol];
            acc[t] = __builtin_amdgcn_wmma_f32_16x16x4_f32(
                false, a, false, b, (short)0, acc[t], false, false);
        }
    }
    // C/D layout: VGPR r -> (M = r, lanes 0-15) and (M = r+8, lanes 16-31); N = lane&15
    int rbase = wave * 16 + ((lane >> 4) << 3);
#pragma unroll
    for (int t = 0; t < 8; ++t)
#pragma unroll
        for (int r = 0; r < 8; ++r)
            C[(long)(rbase + r) * CH + t * 16 + (lane & 15)] = acc[t][r];
}

// ---------------- aggregation ----------------
// agg[b,n,:] = bias[:] + h[b,n,:] * dinv[n]^2   (fuses self-loop edge + bias)
__global__ void gcn_agg_init(const float* __restrict__ h, const float* __restrict__ dinv,
                             const float* __restrict__ bias, float* __restrict__ agg,
                             long total, int n) {
    long i = (long)blockIdx.x * blockDim.x + threadIdx.x;
    if (i >= total) return;
    long row = i >> 7;
    int  c   = (int)(i & 127);
    int  nd  = (int)(row % n);
    float di = dinv[nd];
    agg[i] = bias[c] + h[i] * di * di;
}

// one wave per edge, gridDim.y = batch: float4 gather of 128 channels + 4 HW fp32 atomics/lane
__global__ void gcn_scatter(const float* __restrict__ h, const int* __restrict__ src,
                            const int* __restrict__ dst, const float* __restrict__ dinv,
                            float* __restrict__ agg, int e, int n) {
    int ed = blockIdx.x * (blockDim.x >> 5) + (threadIdx.x >> 5);
    if (ed >= e) return;
    int b = blockIdx.y;
    int s = src[ed], d = dst[ed];
    float nrm = dinv[s] * dinv[d];
    int lane = threadIdx.x & 31;
    const float4* hp = (const float4*)(h + ((long)b * n + s) * CH);
    float*        ap = agg + ((long)b * n + d) * CH + lane * 4;
    float4 v = hp[lane];
    atomicAddF(ap + 0, v.x * nrm);
    atomicAddF(ap + 1, v.y * nrm);
    atomicAddF(ap + 2, v.z * nrm);
    atomicAddF(ap + 3, v.w * nrm);
}

// ---------------- BatchNorm (ReLU fused into the read path) ----------------
// blockDim = 128 (one thread per channel); each block covers 256 rows.
__global__ void gcn_bn_stats(const float* __restrict__ x, float* __restrict__ sums, long rows) {
    int c = threadIdx.x;
    long r0 = (long)blockIdx.x * 256;
    long r1 = r0 + 256; if (r1 > rows) r1 = rows;
    float s = 0.f, q = 0.f;
    for (long r = r0; r < r1; ++r) {
        float v = x[r * CH + c];
        v = fmaxf(v, 0.f);                    // relu before BN stats
        s += v; q += v * v;
    }
    atomicAddF(&sums[c], s);
    atomicAddF(&sums[CH + c], q);
}

__global__ void gcn_bn_apply(const float* __restrict__ x, const float* __restrict__ sums,
                             const float* __restrict__ gamma, const float* __restrict__ beta,
                             float* __restrict__ y, long total, float invcnt) {
    long i = (long)blockIdx.x * blockDim.x + threadIdx.x;
    if (i >= total) return;
    int c = (int)(i & 127);
    float mu  = sums[c] * invcnt;
    float var = sums[CH + c] * invcnt - mu * mu;
    float v = fmaxf(x[i], 0.f);
    y[i] = (v - mu) * rsqrtf(var + BN_EPS) * gamma[c] + beta[c];
}

// ---------------- final: relu + BN2 + classifier dot, one wave per node ----------------
__global__ void gcn_bn_dot_out(const float* __restrict__ x, const float* __restrict__ sums,
                               const float* __restrict__ gamma, const float* __restrict__ beta,
                               const float* __restrict__ wc, const float* __restrict__ bc,
                               float* __restrict__ out, long rows, float invcnt) {
    long wid = (long)blockIdx.x * (blockDim.x >> 5) + (threadIdx.x >> 5);
    if (wid >= rows) return;
    int lane = threadIdx.x & 31;
    const float4* xp = (const float4*)(x + wid * CH);
    float4 v = xp[lane];
    float vv[4] = {v.x, v.y, v.z, v.w};
    float acc = 0.f;
#pragma unroll
    for (int j = 0; j < 4; ++j) {
        int c = lane * 4 + j;
        float mu  = sums[c] * invcnt;
        float var = sums[CH + c] * invcnt - mu * mu;
        float val = (fmaxf(vv[j], 0.f) - mu) * rsqrtf(var + BN_EPS) * gamma[c] + beta[c];
        acc += val * wc[c];
    }
#pragma unroll
    for (int o = 16; o > 0; o >>= 1)
        acc += __shfl_xor(acc, o, 32);        // wave32 reduction
    if (lane == 0) out[wid] = acc + bc[0];
}

extern "C" void kernel_launch(void* const* d_in, const int* in_sizes, int n_in,
                              void* d_out, int out_size, void* d_ws, size_t ws_size,
                              hipStream_t stream) {
    const float* x   = (const float*)d_in[0];
    const int*   ei  = (const int*)d_in[1];
    const float* W1  = (const float*)d_in[2];
    const float* b1  = (const float*)d_in[3];
    const float* W2  = (const float*)d_in[4];
    const float* b2  = (const float*)d_in[5];
    const float* g1  = (const float*)d_in[6];
    const float* be1 = (const float*)d_in[7];
    const float* g2  = (const float*)d_in[8];
    const float* be2 = (const float*)d_in[9];
    const float* Wc  = (const float*)d_in[10];
    const float* bc  = (const float*)d_in[11];
    float* out = (float*)d_out;

    const int  E  = in_sizes[1] / 2;          // 320000
    const int  N  = 20000;
    const int  H  = in_sizes[3];              // 128
    const int  F  = in_sizes[2] / H;          // 64
    const long BN = (long)out_size;           // B*N = 80000
    const int  Bz = (int)(BN / N);            // 4
    const long TOT = BN * H;                  // 10.24M elems
    const float invcnt = 1.0f / (float)BN;

    float* ws     = (float*)d_ws;
    float* stats1 = ws;                       // 256 floats
    float* stats2 = ws + 256;                 // 256 floats
    float* dinv   = ws + 512;                 // N floats
    long   nAl    = (N + 255) & ~255L;
    float* bufA   = ws + 512 + nAl;
    float* bufB   = bufA + TOT;

    const int* srcIdx = ei;
    const int* dstIdx = ei + E;

    // degree + stats init
    gcn_zero     <<<2, 256, 0, stream>>>(stats1, 512);
    gcn_deg_init <<<(N + 255) / 256, 256, 0, stream>>>(dinv, N);
    gcn_deg_count<<<(E + 255) / 256, 256, 0, stream>>>(dstIdx, dinv, E);
    gcn_deg_fin  <<<(N + 255) / 256, 256, 0, stream>>>(dinv, N);

    dim3 edgeGrid((E + 7) / 8, Bz, 1);        // 8 waves per block, y = batch
    int  gemmBlks  = (int)((BN / 16 + 7) / 8);
    int  ewBlks    = (int)((TOT + 255) / 256);
    int  statBlks  = (int)((BN + 255) / 256);
    int  rowBlks   = (int)((BN + 7) / 8);

    // ---- layer 1 ----
    gcn_gemm_wmma<<<gemmBlks, 256, 0, stream>>>(x, W1, bufA, (int)BN, F);
    gcn_agg_init <<<ewBlks, 256, 0, stream>>>(bufA, dinv, b1, bufB, TOT, N);
    gcn_scatter  <<<edgeGrid, 256, 0, stream>>>(bufA, srcIdx, dstIdx, dinv, bufB, E, N);
    gcn_bn_stats <<<statBlks, 128, 0, stream>>>(bufB, stats1, BN);
    gcn_bn_apply <<<ewBlks, 256, 0, stream>>>(bufB, stats1, g1, be1, bufA, TOT, invcnt);

    // ---- layer 2 ----
    gcn_gemm_wmma<<<gemmBlks, 256, 0, stream>>>(bufA, W2, bufB, (int)BN, H);
    gcn_agg_init <<<ewBlks, 256, 0, stream>>>(bufB, dinv, b2, bufA, TOT, N);
    gcn_scatter  <<<edgeGrid, 256, 0, stream>>>(bufB, srcIdx, dstIdx, dinv, bufA, E, N);
    gcn_bn_stats <<<statBlks, 128, 0, stream>>>(bufA, stats2, BN);

    // ---- fused BN2 + classifier ----
    gcn_bn_dot_out<<<rowBlks, 256, 0, stream>>>(bufA, stats2, g2, be2, Wc, bc, out, BN, invcnt);
}